// GedLayer_9844065042951
// MI455X (gfx1250) — compile-verified
//
#include <hip/hip_runtime.h>
#include <hip/hip_bf16.h>
#include <math.h>

// ---------------------------------------------------------------------------
// GED layer for n=m<=111 (reference uses 100), single persistent workgroup.
//   quad = eid*(b1t + b2t) + sum_b <S@E2_b , Kb^T@S>_F      (6 f32 WMMA GEMMs)
//   ged  = 0.5*(quad - sum c*v^2) + sum c*v
// ---------------------------------------------------------------------------

typedef __attribute__((ext_vector_type(2))) float v2f;
typedef __attribute__((ext_vector_type(8))) float v8f;

#define PDIM 112          // 101 padded to 7*16
#define PITCH 113         // +1 to avoid LDS bank conflicts on row-strided loads
#define NTHREADS 256
#define NWAVES 8
#define NTILE 7
#define NTILES 49
#define SK_ITERS 10

#define SMEM_FLOATS (3 * PDIM * PITCH + 100 + 12 + 2 * PDIM + NTHREADS + 2)

__device__ __forceinline__ float relu_f(float x) { return x > 0.f ? x : 0.f; }

__device__ float block_reduce(float x, float* red, int tid) {
  red[tid] = x;
  __syncthreads();
  for (int s = NTHREADS >> 1; s > 0; s >>= 1) {
    if (tid < s) red[tid] += red[tid + s];
    __syncthreads();
  }
  float r = red[0];
  __syncthreads();
  return r;
}

__device__ __forceinline__ float node_cost_at(int i, int k, int n, int m,
                                              const int* l1, const int* l2,
                                              const float* nc, float nid) {
  if (i < n && k < m) return nc[l1[i] * 10 + l2[k]];
  if (i == n && k == m) return 0.f;
  return nid;
}

// C = A @ B, PDIM x PDIM row-major (pitch PITCH), f32 WMMA 16x16x4.
// A frag: lane<16 -> (M=lane, K=k0..k0+1), lane>=16 -> (M=lane-16, K=k0+2..k0+3)
// B frag: lane<16 -> (K=k0..k0+1, N=lane), lane>=16 -> (K=k0+2..k0+3, N=lane-16)
// C/D:    VGPR r -> row r (lanes 0-15) / row r+8 (lanes 16-31)
__device__ void gemm_store(const float* __restrict__ A, const float* __restrict__ B,
                           float* __restrict__ C, int lane, int wave) {
  const int half = lane >> 4, l15 = lane & 15;
  for (int t = wave; t < NTILES; t += NWAVES) {
    const int bm = t / NTILE, bn = t % NTILE;
    v8f acc = {};
    const float* arow = A + (bm * 16 + l15) * PITCH;
    const float* bcol = B + bn * 16 + l15;
    for (int k0 = 0; k0 < PDIM; k0 += 4) {
      const int ka = k0 + half * 2;
      v2f av, bv;
      av[0] = arow[ka];
      av[1] = arow[ka + 1];
      bv[0] = bcol[ka * PITCH];
      bv[1] = bcol[(ka + 1) * PITCH];
      acc = __builtin_amdgcn_wmma_f32_16x16x4_f32(false, av, false, bv, (short)0,
                                                  acc, false, false);
    }
    float* cp = C + (bm * 16 + half * 8) * PITCH + bn * 16 + l15;
#pragma unroll
    for (int r = 0; r < 8; ++r) cp[r * PITCH] = acc[r];
  }
}

// returns per-thread partial of <A@B, M>_F (N never materialized)
__device__ float gemm_dot(const float* __restrict__ A, const float* __restrict__ B,
                          const float* __restrict__ M, int lane, int wave) {
  const int half = lane >> 4, l15 = lane & 15;
  float partial = 0.f;
  for (int t = wave; t < NTILES; t += NWAVES) {
    const int bm = t / NTILE, bn = t % NTILE;
    v8f acc = {};
    const float* arow = A + (bm * 16 + l15) * PITCH;
    const float* bcol = B + bn * 16 + l15;
    for (int k0 = 0; k0 < PDIM; k0 += 4) {
      const int ka = k0 + half * 2;
      v2f av, bv;
      av[0] = arow[ka];
      av[1] = arow[ka + 1];
      bv[0] = bcol[ka * PITCH];
      bv[1] = bcol[(ka + 1) * PITCH];
      acc = __builtin_amdgcn_wmma_f32_16x16x4_f32(false, av, false, bv, (short)0,
                                                  acc, false, false);
    }
    const float* mp = M + (bm * 16 + half * 8) * PITCH + bn * 16 + l15;
#pragma unroll
    for (int r = 0; r < 8; ++r) partial += acc[r] * mp[r * PITCH];
  }
  return partial;
}

__global__ __launch_bounds__(NTHREADS) void ged_kernel(
    const int* __restrict__ A1, const int* __restrict__ l1,
    const int* __restrict__ A2, const int* __restrict__ l2,
    const float* __restrict__ nw, const float* __restrict__ ew,
    const int* __restrict__ np, const int* __restrict__ mp,
    float* __restrict__ out) {
  extern __shared__ float smem[];
  float* S    = smem;                     // PDIM x PITCH
  float* BUF  = S + PDIM * PITCH;         // PDIM x PITCH (E2_b, then Kb^T)
  float* MB   = BUF + PDIM * PITCH;       // PDIM x PITCH (S @ E2_b)
  float* nc   = MB + PDIM * PITCH;        // 10x10 node costs
  float* Ktab = nc + 100;                 // [4][3]
  float* rs   = Ktab + 12;                // row sums
  float* cs   = rs + PDIM;                // col sums
  float* red  = cs + PDIM;                // reduction scratch
  float* scal = red + NTHREADS;           // [0]=node ins/del, [1]=edge ins/del

  const int tid = threadIdx.x;
  const int lane = tid & 31;
  const int wave = tid >> 5;
  const int n = np[0], m = mp[0];
  const int n1 = n + 1, m1 = m + 1;

  if (tid < 64) {  // pull A1/A2 toward caches (global_prefetch_b8)
    __builtin_prefetch(A1 + tid * 160, 0, 3);
    __builtin_prefetch(A2 + tid * 160, 0, 3);
  }

  if (tid == 0) {
    for (int i = 0; i < 100; ++i) nc[i] = 0.f;
    int p = 0;
    for (int r = 0; r < 10; ++r)
      for (int c = r + 1; c < 10; ++c) {
        float w = relu_f(nw[p]); nc[r * 10 + c] = w; nc[c * 10 + r] = w; ++p;
      }
    scal[0] = relu_f(nw[p]);  // p == 45
    float ec[9];
    for (int i = 0; i < 9; ++i) ec[i] = 0.f;
    p = 0;
    for (int r = 0; r < 3; ++r)
      for (int c = r + 1; c < 3; ++c) {
        float w = relu_f(ew[p]); ec[r * 3 + c] = w; ec[c * 3 + r] = w; ++p;
      }
    float eidv = relu_f(ew[p]);  // p == 3
    scal[1] = eidv;
    for (int b = 0; b < 3; ++b) Ktab[b] = 0.f;
    for (int q = 1; q <= 3; ++q)
      for (int b = 0; b < 3; ++b)
        Ktab[q * 3 + b] = ec[(q - 1) * 3 + b] - 2.f * eidv;
  }
  __syncthreads();
  const float nid = scal[0];
  const float eid = scal[1];

  // S0 = exp(-0.5 * c), zero-padded
  for (int e = tid; e < PDIM * PDIM; e += NTHREADS) {
    int i = e / PDIM, k = e % PDIM;
    float v = 0.f;
    if (i < n1 && k < m1)
      v = expf(-0.5f * node_cost_at(i, k, n, m, l1, l2, nc, nid));
    S[i * PITCH + k] = v;
  }
  __syncthreads();

  // Sinkhorn (last row / last col not normalized)
  for (int it = 0; it < SK_ITERS; ++it) {
    if (tid < n) {
      float s = 0.f;
      for (int k = 0; k < m1; ++k) s += S[tid * PITCH + k];
      float inv = 1.f / s;
      for (int k = 0; k < m1; ++k) S[tid * PITCH + k] *= inv;
    }
    __syncthreads();
    if (tid < m) {
      float s = 0.f;
      for (int i = 0; i < n1; ++i) s += S[i * PITCH + tid];
      float inv = 1.f / s;
      for (int i = 0; i < n1; ++i) S[i * PITCH + tid] *= inv;
    }
    __syncthreads();
  }

  if (tid < n1) { float s = 0.f; for (int k = 0; k < m1; ++k) s += S[tid * PITCH + k]; rs[tid] = s; }
  if (tid < m1) { float s = 0.f; for (int i = 0; i < n1; ++i) s += S[i * PITCH + tid]; cs[tid] = s; }
  __syncthreads();

  float pb1 = 0.f, pb2 = 0.f, pcv = 0.f, pcv2 = 0.f;
  for (int e = tid; e < n * n; e += NTHREADS)
    if (A1[e] != 0) pb1 += rs[e / n] * rs[e % n];
  for (int e = tid; e < m * m; e += NTHREADS)
    if (A2[e] != 0) pb2 += cs[e / m] * cs[e % m];
  for (int e = tid; e < n1 * m1; e += NTHREADS) {
    int i = e / m1, k = e % m1;
    float cv = node_cost_at(i, k, n, m, l1, l2, nc, nid);
    float v = S[i * PITCH + k];
    pcv += cv * v;
    pcv2 += cv * v * v;
  }

  float pequad = 0.f;
  for (int b = 1; b <= 3; ++b) {
    // BUF = E2_b (indicator A2==b, zero-padded)
    for (int e = tid; e < PDIM * PDIM; e += NTHREADS) {
      int k = e / PDIM, l = e % PDIM;
      BUF[k * PITCH + l] = (k < m && l < m && A2[k * m + l] == b) ? 1.f : 0.f;
    }
    __syncthreads();
    gemm_store(S, BUF, MB, lane, wave);  // MB = S @ E2_b
    __syncthreads();
    // BUF[j][i] = K[A1pad(i,j)][b-1]  (Kb^T)
    for (int e = tid; e < PDIM * PDIM; e += NTHREADS) {
      int j = e / PDIM, i = e % PDIM;
      int lab = (i < n && j < n) ? A1[i * n + j] : 0;
      BUF[j * PITCH + i] = Ktab[lab * 3 + (b - 1)];
    }
    __syncthreads();
    pequad += gemm_dot(BUF, S, MB, lane, wave);  // <MB, Kb^T @ S>_F
    __syncthreads();
  }

  float b1t  = block_reduce(pb1, red, tid);
  float b2t  = block_reduce(pb2, red, tid);
  float cvt  = block_reduce(pcv, red, tid);
  float cv2t = block_reduce(pcv2, red, tid);
  float eqd  = block_reduce(pequad, red, tid);

  if (tid == 0) {
    float quad = eid * (b1t + b2t) + eqd;
    out[0] = 0.5f * (quad - cv2t) + cvt;
  }
}

extern "C" void kernel_launch(void* const* d_in, const int* in_sizes, int n_in,
                              void* d_out, int out_size, void* d_ws, size_t ws_size,
                              hipStream_t stream) {
  (void)in_sizes; (void)n_in; (void)out_size; (void)d_ws; (void)ws_size;
  const int*   A1 = (const int*)d_in[0];
  const int*   l1 = (const int*)d_in[1];
  const int*   A2 = (const int*)d_in[2];
  const int*   l2 = (const int*)d_in[3];
  const float* nw = (const float*)d_in[4];
  const float* ew = (const float*)d_in[5];
  const int*   np = (const int*)d_in[6];
  const int*   mp = (const int*)d_in[7];
  float* out = (float*)d_out;

  const size_t smem = SMEM_FLOATS * sizeof(float);  // ~151 KB of the WGP's 320 KB LDS
  hipFuncSetAttribute((const void*)ged_kernel,
                      hipFuncAttributeMaxDynamicSharedMemorySize, (int)smem);
  ged_kernel<<<dim3(1), dim3(NTHREADS), smem, stream>>>(A1, l1, A2, l2, nw, ew, np, mp, out);
}